// Curating_of_attention_loss_18047452578241
// MI455X (gfx1250) — compile-verified
//
#include <hip/hip_runtime.h>
#include <hip/hip_bf16.h>
#include <stddef.h>

// ---------------------------------------------------------------------------
// Curating-of-attention loss on MI455X (gfx1250, wave32).
//   out = sum(sattn * pen),  pen[b,i,j] = pf_bug[b, gf[i], gf[j]]
//   D[p,q] = (n_p + n_q - 2*<G_p,G_q>)/9,  M = Ghat * Ghat^T via V_WMMA_F32_16X16X4_F32
// Memory-bound: sattn = 256 MB streamed once (~11 us @ 23.3 TB/s).
// Scratch D (16 MB) stays resident in the 192 MB L2 between kernels.
// ---------------------------------------------------------------------------

typedef float v2f __attribute__((ext_vector_type(2)));
typedef float v8f __attribute__((ext_vector_type(8)));

#define NIMG 64
#define NP   256           // patches per image (16x16)
#define NPIX 1024          // 32*32

// ---------------------------------------------------------------------------
// Kernel 1: per image, Gram features -> M = Ghat*Ghat^T (WMMA) -> D + row stats
// ---------------------------------------------------------------------------
__global__ __launch_bounds__(256)
void k_gram_wmma(const float* __restrict__ batch,
                 float* __restrict__ wsD,      // [64][256][256]
                 float* __restrict__ stats) {  // [64][256][2] = {min,max} of D row
  __shared__ float Gh[NP][12];   // flattened 3x3 Gram per patch, K padded to 12
  __shared__ float nrm[NP];      // ||G_p||^2 (sum of 9 squares)

  const int b   = blockIdx.x;
  const int tid = threadIdx.x;

  // ---- build Ghat: thread == patch ----
  {
    const int p  = tid;
    const int py = p >> 4, px = p & 15;
    float v[3][4];
#pragma unroll
    for (int c = 0; c < 3; ++c)
#pragma unroll
      for (int f = 0; f < 4; ++f) {
        const int y = 2 * py + (f >> 1);
        const int x = 2 * px + (f & 1);
        v[c][f] = batch[(((size_t)b * 3 + c) * 32 + y) * 32 + x];
      }
    float n = 0.0f;
#pragma unroll
    for (int i = 0; i < 3; ++i)
#pragma unroll
      for (int j = 0; j < 3; ++j) {
        float s = 0.0f;
#pragma unroll
        for (int f = 0; f < 4; ++f) s += v[i][f] * v[j][f];
        s *= (1.0f / 12.0f);            // / (3 * GRID_L * GRID_L)
        Gh[p][i * 3 + j] = s;
        n += s * s;
      }
    Gh[p][9] = 0.0f; Gh[p][10] = 0.0f; Gh[p][11] = 0.0f;  // K padding
    nrm[p] = n;
  }
  __syncthreads();

  // ---- WMMA tiles: M[p,q] = sum_k Ghat[p][k]*Ghat[q][k], K = 12 (3 slices of 4)
  const int wave = tid >> 5;
  const int lane = tid & 31;
  const int m    = lane & 15;
  const int kb   = (lane < 16) ? 0 : 2;   // 16x4 f32 A layout: VGPR v -> K = kb+v
  const int rsel = (lane < 16) ? 0 : 8;   // 16x16 f32 C layout

  for (int strip = 0; strip < 2; ++strip) {
    const int ar0 = (wave + strip * 8) * 16;   // tile-row base (rows ar0..ar0+15)
    float rmin[8], rmax[8];
#pragma unroll
    for (int r = 0; r < 8; ++r) { rmin[r] = 3.4e38f; rmax[r] = -3.4e38f; }

    for (int tn = 0; tn < 16; ++tn) {
      const int bc0 = tn * 16;
      v8f c = {};
#pragma unroll
      for (int s = 0; s < 3; ++s) {
        v2f a, bb;
        a.x  = Gh[ar0 + m][4 * s + kb];
        a.y  = Gh[ar0 + m][4 * s + kb + 1];
        bb.x = Gh[bc0 + m][4 * s + kb];     // B[k][n] = Ghat[bc0+n][k] (symmetric)
        bb.y = Gh[bc0 + m][4 * s + kb + 1];
        c = __builtin_amdgcn_wmma_f32_16x16x4_f32(
                false, a, false, bb, (short)0, c, false, false);
      }
      const int   col  = bc0 + m;
      const float ncol = nrm[col];
#pragma unroll
      for (int r = 0; r < 8; ++r) {
        const int row = ar0 + rsel + r;
        const float d = (nrm[row] + ncol - 2.0f * c[r]) * (1.0f / 9.0f);
        wsD[(((size_t)b * NP + row) << 8) + col] = d;
        rmin[r] = fminf(rmin[r], d);
        rmax[r] = fmaxf(rmax[r], d);
      }
    }
    // reduce min/max across the 16 lanes sharing each row (masks<16 keep halves apart)
#pragma unroll
    for (int r = 0; r < 8; ++r) {
#pragma unroll
      for (int msk = 1; msk < 16; msk <<= 1) {
        rmin[r] = fminf(rmin[r], __shfl_xor(rmin[r], msk, 32));
        rmax[r] = fmaxf(rmax[r], __shfl_xor(rmax[r], msk, 32));
      }
    }
    if (m == 0) {
#pragma unroll
      for (int r = 0; r < 8; ++r) {
        const int row = ar0 + rsel + r;
        stats[(((size_t)b * NP + row) << 1)    ] = rmin[r];
        stats[(((size_t)b * NP + row) << 1) + 1] = rmax[r];
      }
    }
  }
}

// ---------------------------------------------------------------------------
// Kernel 2: one block per (image b, patch-row p). Normalize+bug pf row into
// LDS, stream the 4 matching sattn rows (4 KB each) and accumulate.
// ---------------------------------------------------------------------------
__global__ __launch_bounds__(256)
void k_weighted_sum(const float* __restrict__ sattn,
                    const float* __restrict__ wsD,
                    const float* __restrict__ stats,
                    float* __restrict__ partial) {
  __shared__ float w256[NP];
  __shared__ float red[256];

  const int blk = blockIdx.x;
  const int b   = blk >> 8;
  const int p   = blk & 255;
  const int tid = threadIdx.x;

  const size_t rowbase = ((size_t)b * NP + p) << 8;
  const float  mn  = stats[(((size_t)b * NP + p) << 1)];
  const float  mx  = stats[(((size_t)b * NP + p) << 1) + 1];
  const float  inv = 1.0f / (mx - mn);

  {   // pf row, with the emulated gather "bug" on column 1
    const int q = tid;
    float pf = (wsD[rowbase + q] - mn) * inv + 1.0f;
    if (q == 1) {
      const float pf2 = (wsD[rowbase + 2] - mn) * inv + 1.0f;
      if (pf == 2.0f) pf = pf2;   // pf[p,1]==2.0 <=> col 1 is the row max
    }
    w256[q] = pf;
  }
  __syncthreads();

  // 4 sattn rows i = (2*py+dy)*32 + (2*px+dx); weight depends only on j.
  const int py = p >> 4, px = p & 15;
  const int j  = tid * 4;                 // one float4 per thread per row
  const int y2 = j >> 5, x2 = j & 31;
  const int g0 = (y2 >> 1) * 16 + (x2 >> 1);
  const float w0 = w256[g0];
  const float w1 = w256[g0 + 1];

  const size_t sbase = ((size_t)b << 20);
  float sx = 0.0f, sy = 0.0f, sz = 0.0f, sw = 0.0f;
#pragma unroll
  for (int dy = 0; dy < 2; ++dy)
#pragma unroll
    for (int dx = 0; dx < 2; ++dx) {
      const int i = (2 * py + dy) * 32 + (2 * px + dx);
      const float4 v = *(const float4*)(sattn + sbase + ((size_t)i << 10) + j);
      sx += v.x; sy += v.y; sz += v.z; sw += v.w;
    }
  float acc = (sx + sy) * w0 + (sz + sw) * w1;

  red[tid] = acc;
  __syncthreads();
#pragma unroll
  for (int s = 128; s > 0; s >>= 1) {
    if (tid < s) red[tid] += red[tid + s];
    __syncthreads();
  }
  if (tid == 0) partial[blk] = red[0];
}

// ---------------------------------------------------------------------------
// Kernel 3: deterministic final reduction of 16384 partials -> scalar
// ---------------------------------------------------------------------------
__global__ __launch_bounds__(256)
void k_final_reduce(const float* __restrict__ partial, float* __restrict__ out) {
  __shared__ float red[256];
  float acc = 0.0f;
  for (int i = threadIdx.x; i < NIMG * NP; i += 256) acc += partial[i];
  red[threadIdx.x] = acc;
  __syncthreads();
#pragma unroll
  for (int s = 128; s > 0; s >>= 1) {
    if (threadIdx.x < s) red[threadIdx.x] += red[threadIdx.x + s];
    __syncthreads();
  }
  if (threadIdx.x == 0) out[0] = red[0];
}

// ---------------------------------------------------------------------------
extern "C" void kernel_launch(void* const* d_in, const int* in_sizes, int n_in,
                              void* d_out, int out_size, void* d_ws, size_t ws_size,
                              hipStream_t stream) {
  (void)in_sizes; (void)n_in; (void)out_size; (void)ws_size;
  const float* batch = (const float*)d_in[0];   // (64,3,32,32) f32
  const float* sattn = (const float*)d_in[1];   // (64,32,32,32,32) f32
  float* out = (float*)d_out;                   // scalar f32

  char* ws = (char*)d_ws;
  const size_t szD  = (size_t)NIMG * NP * NP * sizeof(float);   // 16 MB
  const size_t szS  = (size_t)NIMG * NP * 2 * sizeof(float);    // 128 KB
  float* wsD     = (float*)ws;
  float* stats   = (float*)(ws + szD);
  float* partial = (float*)(ws + szD + szS);                    // 16384 floats

  k_gram_wmma   <<<NIMG,        256, 0, stream>>>(batch, wsD, stats);
  k_weighted_sum<<<NIMG * NP,   256, 0, stream>>>(sattn, wsD, stats, partial);
  k_final_reduce<<<1,           256, 0, stream>>>(partial, out);
}